// Polynomial_55843164782886
// MI455X (gfx1250) — compile-verified
//
#include <hip/hip_runtime.h>

#define BATCH 32768
#define IN_DIM 128
#define OUT_DIM 64
#define NFEAT 8384            // original features: 128 + 128*129/2
#define NFP   8576            // padded: 128 + sum_i ceil((128-i)/4)*4 = 134 panels of 64
#define NCHUNK (NFP / 4)      // 2144 K-chunks of 4 features
#define KP 64                 // k-panel size
#define NPANEL (NFP / KP)     // 134
#define ROWS_PER_BLOCK 128
#define XST 132               // xs row stride: 128 x + ones col + 3 zero pad cols
#define WST 76                // wbuf row stride (bank skew, 8B-aligned pairs)

typedef float v2f __attribute__((ext_vector_type(2)));
typedef float v8f __attribute__((ext_vector_type(8)));
typedef int   vi4 __attribute__((__vector_size__(16)));   // matches builtin param type

#ifndef TRY_ASYNC
#define TRY_ASYNC 1
#endif
#if TRY_ASYNC && __has_builtin(__builtin_amdgcn_global_load_async_to_lds_b128)
#define HAVE_ASYNC 1
#else
#define HAVE_ASYNC 0
#endif

__device__ __forceinline__ void wait_async0() {
#if __has_builtin(__builtin_amdgcn_s_wait_asynccnt)
  __builtin_amdgcn_s_wait_asynccnt(0);
#else
  asm volatile("s_wait_asynccnt 0" ::: "memory");
#endif
}

#if HAVE_ASYNC
__device__ __forceinline__ void async_cp16(const void* g, void* l) {
  __builtin_amdgcn_global_load_async_to_lds_b128(
      (__attribute__((address_space(1))) vi4*)g,
      (__attribute__((address_space(3))) vi4*)l, 0, 0);
}
#endif

// ---- setup kernel A: per-4-chunk metadata (i, j0); i==128 means "ones" (deg-1) ----
__global__ __launch_bounds__(256) void build_meta(int* __restrict__ meta) {
  int c = blockIdx.x * 256 + threadIdx.x;
  if (c >= NCHUNK) return;
  int i, j0;
  if (c < IN_DIM / 4) {            // degree-1 chunks: A = 1 * x[j0..j0+3]
    i = IN_DIM; j0 = c * 4;
  } else {
    int rem = c - IN_DIM / 4;
    i = 0;
    for (; i < IN_DIM; ++i) {
      int gc = (IN_DIM - i + 3) >> 2;     // padded chunks in group i
      if (rem < gc) break;
      rem -= gc;
    }
    j0 = i + rem * 4;
  }
  meta[c] = (i << 16) | j0;
}

// ---- setup kernel B: padded, zero-filled repack Wp[o][kpad] ----
__global__ __launch_bounds__(256) void build_wp(const float* __restrict__ W,
                                                const int* __restrict__ meta,
                                                float* __restrict__ Wp) {
  int idx = blockIdx.x * 256 + threadIdx.x;
  if (idx >= NFP * OUT_DIM) return;
  int o = idx / NFP;
  int kp = idx - o * NFP;
  int m = meta[kp >> 2];
  int i = m >> 16;
  int j = (m & 0xffff) + (kp & 3);
  float v = 0.0f;
  if (i == IN_DIM) {                       // degree-1 feature j
    v = W[(size_t)o * NFEAT + j];
  } else if (j < IN_DIM) {                 // degree-2 feature (i,j), flat k
    int k = IN_DIM + i * (257 - i) / 2 + (j - i);
    v = W[(size_t)o * NFEAT + k];
  }                                        // else: pad slot -> 0
  Wp[idx] = v;
}

// ---- main kernel: on-the-fly polynomial-feature GEMM via f32 WMMA ----
__global__ __launch_bounds__(256) void poly_wmma_f32(
    const float* __restrict__ x, const float* __restrict__ Wp,
    const float* __restrict__ bias, const int* __restrict__ meta,
    float* __restrict__ out)
{
  __shared__ __align__(16) float xs[ROWS_PER_BLOCK * XST];
  __shared__ __align__(16) float wbuf[OUT_DIM * WST];   // [o][kk]
  __shared__ int mbuf[KP / 4];

  const int tid   = threadIdx.x;
  const int lane  = tid & 31;
  const int wave  = tid >> 5;
  const int half  = lane >> 4;
  const int half2 = half * 2;
  const int n     = lane & 15;
  const int row0  = blockIdx.x * ROWS_PER_BLOCK;

  // ---- stage x tile (128x128 f32) into LDS ----
  {
    const float4* xg = (const float4*)(x + (size_t)row0 * IN_DIM);
#pragma unroll
    for (int it = 0; it < 16; ++it) {
      int e = tid + it * 256;              // 4096 float4
      int r = e >> 5;
      int c = (e & 31) << 2;
#if HAVE_ASYNC
      async_cp16(&xg[(size_t)r * 32 + (e & 31)], &xs[r * XST + c]);
#else
      float4 v = xg[(size_t)r * 32 + (e & 31)];
      float* d = &xs[r * XST + c];
      d[0] = v.x; d[1] = v.y; d[2] = v.z; d[3] = v.w;
#endif
    }
#if HAVE_ASYNC
    wait_async0();
#endif
    if (tid < ROWS_PER_BLOCK) {            // ones col + zero pad (pads hit W==0)
      float* d = &xs[tid * XST + IN_DIM];
      d[0] = 1.0f; d[1] = 0.0f; d[2] = 0.0f; d[3] = 0.0f;
    }
  }

  v8f acc0 = {}, acc1 = {}, acc2 = {}, acc3 = {};
  const int rowbase = (wave * 16 + n) * XST;   // A row M = lane&15

  // register double-buffer for W panel + metadata
  const int o_st  = tid >> 2;
  const int kk_st = (tid & 3) << 4;            // 0,16,32,48
  float4 r0, r1, r2, r3;
  int mreg = 0;
  {
    const float4* wr = (const float4*)(Wp + (size_t)o_st * NFP + kk_st);
    r0 = wr[0]; r1 = wr[1]; r2 = wr[2]; r3 = wr[3];
    if (tid < KP / 4) mreg = meta[tid];
  }

  for (int p = 0; p < NPANEL; ++p) {
    __syncthreads();                           // prior compute done
    {
      float4* dst = (float4*)&wbuf[o_st * WST + kk_st];
      dst[0] = r0; dst[1] = r1; dst[2] = r2; dst[3] = r3;
      if (tid < KP / 4) mbuf[tid] = mreg;
    }
    __syncthreads();
    if (p + 1 < NPANEL) {                      // prefetch next panel (hidden by compute)
      const float4* wr = (const float4*)(Wp + (size_t)o_st * NFP + (p + 1) * KP + kk_st);
      r0 = wr[0]; r1 = wr[1]; r2 = wr[2]; r3 = wr[3];
      if (tid < KP / 4) mreg = meta[(p + 1) * (KP / 4) + tid];
    }

#pragma unroll
    for (int ks = 0; ks < KP / 4; ++ks) {
      const int m  = __builtin_amdgcn_readfirstlane(mbuf[ks]);
      const int i  = m >> 16;
      const int j0 = m & 0xffff;
      const int krow = ks * 4 + half2;         // K base for this lane half

      // A fragment: x_i * x[j0+half2 .. +1]  (4-chunk lies inside one group)
      const float xi = xs[rowbase + i];
      const float xa = xs[rowbase + j0 + half2];
      const float xb = xs[rowbase + j0 + half2 + 1];
      v2f Af; Af[0] = xi * xa; Af[1] = xi * xb;

      // B fragments: contiguous K-pair per lane from o-major panel
      float2 b0 = *(const float2*)&wbuf[(n     ) * WST + krow];
      float2 b1 = *(const float2*)&wbuf[(n + 16) * WST + krow];
      float2 b2 = *(const float2*)&wbuf[(n + 32) * WST + krow];
      float2 b3 = *(const float2*)&wbuf[(n + 48) * WST + krow];
      v2f B0; B0[0] = b0.x; B0[1] = b0.y;
      v2f B1; B1[0] = b1.x; B1[1] = b1.y;
      v2f B2; B2[0] = b2.x; B2[1] = b2.y;
      v2f B3; B3[0] = b3.x; B3[1] = b3.y;

      acc0 = __builtin_amdgcn_wmma_f32_16x16x4_f32(false, Af, false, B0, (short)0, acc0, false, false);
      acc1 = __builtin_amdgcn_wmma_f32_16x16x4_f32(false, Af, false, B1, (short)0, acc1, false, false);
      acc2 = __builtin_amdgcn_wmma_f32_16x16x4_f32(false, Af, false, B2, (short)0, acc2, false, false);
      acc3 = __builtin_amdgcn_wmma_f32_16x16x4_f32(false, Af, false, B3, (short)0, acc3, false, false);
    }
  }

  // ---- epilogue: D VGPR v holds row M = v + 8*half, col N = t*16 + n ----
  const int gRow0 = row0 + wave * 16 + half * 8;
  const float b0v = bias[n];
  const float b1v = bias[16 + n];
  const float b2v = bias[32 + n];
  const float b3v = bias[48 + n];
#pragma unroll
  for (int v = 0; v < 8; ++v) {
    float* o = out + (size_t)(gRow0 + v) * OUT_DIM;
    o[n]      = acc0[v] + b0v;
    o[16 + n] = acc1[v] + b1v;
    o[32 + n] = acc2[v] + b2v;
    o[48 + n] = acc3[v] + b3v;
  }
}

extern "C" void kernel_launch(void* const* d_in, const int* in_sizes, int n_in,
                              void* d_out, int out_size, void* d_ws, size_t ws_size,
                              hipStream_t stream) {
  const float* x = (const float*)d_in[0];
  const float* W = (const float*)d_in[1];
  const float* b = (const float*)d_in[2];
  float* out = (float*)d_out;

  // workspace layout: Wp (NFP*64 f32 = 2.195 MB) | meta (2144 int)
  float* Wp  = (float*)d_ws;
  int*  meta = (int*)((char*)d_ws + (size_t)NFP * OUT_DIM * sizeof(float));

  build_meta<<<(NCHUNK + 255) / 256, 256, 0, stream>>>(meta);
  build_wp<<<(NFP * OUT_DIM + 255) / 256, 256, 0, stream>>>(W, meta, Wp);

  dim3 grid(BATCH / ROWS_PER_BLOCK);   // 256 blocks
  poly_wmma_f32<<<grid, 256, 0, stream>>>(x, Wp, b, meta, out);
}